// radon_14035953123378
// MI455X (gfx1250) — compile-verified
//
#include <hip/hip_runtime.h>

#define N_ANGLES 512
#define IMG 256
#define ANG_PER_BLOCK 4
#define THREADS 512
#define PI_D 3.14159265358979323846

typedef float v2f __attribute__((ext_vector_type(2)));
typedef float v8f __attribute__((ext_vector_type(8)));

#if defined(__gfx1250__) && __has_builtin(__builtin_amdgcn_wmma_f32_16x16x4_f32)
#define USE_WMMA 1
#else
#define USE_WMMA 0
#endif

// Bilinear sample from the LDS-resident image with zero-weight for out-of-range
// corners (matches the reference's (w * valid) masking exactly).
__device__ __forceinline__ float bilin(const float* __restrict__ im, float ix, float iy) {
    float x0f = floorf(ix), y0f = floorf(iy);
    float wx = ix - x0f, wy = iy - y0f;
    int x0 = (int)x0f, y0 = (int)y0f;
    int x1 = x0 + 1, y1 = y0 + 1;
    float vx0 = ((unsigned)x0 < 256u) ? 1.f : 0.f;
    float vx1 = ((unsigned)x1 < 256u) ? 1.f : 0.f;
    float vy0 = ((unsigned)y0 < 256u) ? 1.f : 0.f;
    float vy1 = ((unsigned)y1 < 256u) ? 1.f : 0.f;
    int cx0 = min(max(x0, 0), 255), cx1 = min(max(x1, 0), 255);
    int cy0 = min(max(y0, 0), 255), cy1 = min(max(y1, 0), 255);
    const float* r0 = im + (cy0 << 8);
    const float* r1 = im + (cy1 << 8);
    float v00 = r0[cx0], v01 = r0[cx1], v10 = r1[cx0], v11 = r1[cx1];
    float w00 = (1.f - wx) * (1.f - wy) * vx0 * vy0;
    float w01 = wx * (1.f - wy) * vx1 * vy0;
    float w10 = (1.f - wx) * wy * vx0 * vy1;
    float w11 = wx * wy * vx1 * vy1;
    return w00 * v00 + w01 * v01 + w10 * v10 + w11 * v11;
}

__global__ __launch_bounds__(THREADS) void radon_kernel(const float* __restrict__ image,
                                                        float* __restrict__ out) {
    __shared__ float img[IMG * IMG];  // 256 KB: fits one image in the 320 KB WGP LDS

    const int blocksPerBatch = N_ANGLES / ANG_PER_BLOCK;  // 128
    const int b = blockIdx.x / blocksPerBatch;
    const int aBase = (blockIdx.x % blocksPerBatch) * ANG_PER_BLOCK;
    const int t = threadIdx.x;
    const float* src = image + (size_t)b * IMG * IMG;

    // ---- Stage the full image into LDS via CDNA5 async global->LDS DMA ----
#if defined(__gfx1250__)
    {
        unsigned lbase = (unsigned)(size_t)(&img[0]);  // wave-relative LDS byte offset
        unsigned long long gbase = (unsigned long long)(size_t)src;
        // 256 KB / (512 threads * 16 B) = 32 async b128 copies per thread
        for (int p = 0; p < (IMG * IMG * 4) / (THREADS * 16); ++p) {
            unsigned off = (unsigned)(p * (THREADS * 16) + t * 16);
            unsigned laddr = lbase + off;
            asm volatile("global_load_async_to_lds_b128 %0, %1, %2"
                         :
                         : "v"(laddr), "v"(off), "s"(gbase)
                         : "memory");
        }
        asm volatile("s_wait_asynccnt 0" ::: "memory");
    }
#else
    for (int p = t; p < (IMG * IMG) / 4; p += THREADS)
        ((float4*)img)[p] = ((const float4*)src)[p];
#endif
    __syncthreads();

    const float stp = 2.0f / 255.0f;  // linspace(-1,1,256) step

    for (int aa = 0; aa < ANG_PER_BLOCK; ++aa) {
        const int a = aBase + aa;
        const double th = (double)a * (PI_D / 512.0);
        const float c = (float)cos(th);
        const float s = (float)sin(th);

#if USE_WMMA
        // Wave w owns j in [16w, 16w+16). Lane layout follows the 16x4 f32
        // A-matrix spec: M = lane&15, K pair = 2*(lane>>4) + {0,1}.
        const int wave = t >> 5;
        const int lane = t & 31;
        const int m = lane & 15;
        const int hi = lane >> 4;
        const int j = (wave << 4) + m;
        const float lj = -1.0f + j * stp;
        const float px = c * lj;   // gx = px + s*li
        const float qy = -s * lj;  // gy = qy + c*li

        v8f acc = {};
        const v2f ones = {1.0f, 1.0f};  // B = 4x16 all-ones -> D rows = running ray sums
        for (int stepi = 0; stepi < 64; ++stepi) {
            const int i0 = (stepi << 2) + (hi << 1);
            const float li0 = -1.0f + i0 * stp;
            const float li1 = li0 + stp;
            v2f av;
            {
                float gx = fmaf(s, li0, px);
                float gy = fmaf(c, li0, qy);
                av.x = bilin(img, fmaf(gx, 127.5f, 127.5f), fmaf(gy, 127.5f, 127.5f));
                gx = fmaf(s, li1, px);
                gy = fmaf(c, li1, qy);
                av.y = bilin(img, fmaf(gx, 127.5f, 127.5f), fmaf(gy, 127.5f, 127.5f));
            }
            acc = __builtin_amdgcn_wmma_f32_16x16x4_f32(
                /*neg_a=*/false, av, /*neg_b=*/false, ones,
                /*c_mod=*/(short)0, acc, /*reuse_a=*/false, /*reuse_b=*/false);
        }
        // D layout: VGPR r, lanes 0-15 -> M=r ; lanes 16-31 -> M=8+r (columns replicated).
        if (m == 0) {
            const int jb = (wave << 4) + (hi << 3);
            float* o = out + ((size_t)b * IMG + jb) * N_ANGLES + a;
#pragma unroll
            for (int r = 0; r < 8; ++r) o[(size_t)r * N_ANGLES] = acc[r];
        }
#else
        if (t < IMG) {
            const int j = t;
            const float lj = -1.0f + j * stp;
            const float px = c * lj;
            const float qy = -s * lj;
            float acc = 0.0f;
            for (int i = 0; i < IMG; ++i) {
                const float li = -1.0f + i * stp;
                const float gx = fmaf(s, li, px);
                const float gy = fmaf(c, li, qy);
                acc += bilin(img, fmaf(gx, 127.5f, 127.5f), fmaf(gy, 127.5f, 127.5f));
            }
            out[((size_t)b * IMG + j) * N_ANGLES + a] = acc;
        }
#endif
    }
}

extern "C" void kernel_launch(void* const* d_in, const int* in_sizes, int n_in,
                              void* d_out, int out_size, void* d_ws, size_t ws_size,
                              hipStream_t stream) {
    (void)in_sizes; (void)n_in; (void)out_size; (void)d_ws; (void)ws_size;
    const float* image = (const float*)d_in[0];
    float* out = (float*)d_out;
    dim3 grid(2 * (N_ANGLES / ANG_PER_BLOCK));  // 256 blocks: (batch, 4-angle chunk)
    radon_kernel<<<grid, THREADS, 0, stream>>>(image, out);
}